// RNN_55843164783157
// MI455X (gfx1250) — compile-verified
//
#include <hip/hip_runtime.h>
#include <hip/hip_bf16.h>
#include <math.h>

// ---------------------------------------------------------------------------
// Sizes fixed by the reference: S (sequence = B) = 1024, N = 1024, T = 8, L = 2.
// ---------------------------------------------------------------------------
#define SEQ  1024
#define NN   1024
#define TT   8
#define LL   2

typedef float v2f __attribute__((ext_vector_type(2)));
typedef float v8f __attribute__((ext_vector_type(8)));

// ---------------------------------------------------------------------------
// seqA[s][n] = x[s][0][n]
// ---------------------------------------------------------------------------
__global__ __launch_bounds__(256) void rnn_extract_x(const float* __restrict__ x,
                                                     float* __restrict__ seq) {
    int idx = blockIdx.x * 256 + threadIdx.x;          // 0 .. S*N-1
    int s = idx >> 10;
    int n = idx & (NN - 1);
    seq[idx] = x[(size_t)s * (TT * NN) + n];           // x[s, 0, n]
}

// ---------------------------------------------------------------------------
// out[s][t][n] = src[s][n]
// ---------------------------------------------------------------------------
__global__ __launch_bounds__(256) void rnn_store_out(const float* __restrict__ src,
                                                     float* __restrict__ out, int t) {
    int idx = blockIdx.x * 256 + threadIdx.x;
    int s = idx >> 10;
    int n = idx & (NN - 1);
    out[(size_t)s * (TT * NN) + (size_t)t * NN + n] = src[idx];
}

// ---------------------------------------------------------------------------
// C[M][N] = A[M][K] @ W[N][K]^T + b1[N] (+ b2[N])      M = K = N = 1024, fp32.
//
// Uses V_WMMA_F32_16X16X4_F32 (wave32, exact fp32 — matches reference
// precision; this workload is latency/BW bound so low precision buys nothing).
//
// Per the CDNA5 ISA VGPR layouts:
//   A 16x4 frag : lane l -> row m0+(l&15), cols k+2*(l>>4)+{0,1}  -> float2 load
//   B 4x16 frag : B[k][n] = W[n][k]; same mirrored pattern -> identical float2
//                 load from row-major W: row n0+(l&15), cols k+2*(l>>4)+{0,1}
//   D 16x16 f32 : acc[j] -> row m0 + j + 8*(l>>4), col n0+(l&15)
// Bias is per-column -> identical across acc[0..7]; fold into acc init.
// ---------------------------------------------------------------------------
__global__ __launch_bounds__(256) void rnn_gemm_wmma_bias(
        const float* __restrict__ A,    // [1024][1024] row-major
        const float* __restrict__ W,    // [1024][1024] row-major ([out][in])
        const float* __restrict__ b1,   // [1024]
        const float* __restrict__ b2,   // [1024] or nullptr
        float* __restrict__ C)          // [1024][1024]
{
    const int K = NN;
    int lane = threadIdx.x & 31;
    int wave = threadIdx.x >> 5;
    int lm   = lane & 15;
    int lh   = lane >> 4;
    int n0   = (blockIdx.x * 8 + wave) * 16;   // 8 waves/WG tile the N dim
    int m0   = blockIdx.y * 16;

    float bias = b1[n0 + lm];
    if (b2) bias += b2[n0 + lm];
    v8f acc;
#pragma unroll
    for (int j = 0; j < 8; ++j) acc[j] = bias;

    const float* aptr = A + (size_t)(m0 + lm) * K + 2 * lh;
    const float* wptr = W + (size_t)(n0 + lm) * K + 2 * lh;

#pragma unroll 8
    for (int k = 0; k < K; k += 4) {
        if ((k & 255) == 0) {                   // stream-ahead: global_prefetch_b8
            __builtin_prefetch(aptr + k + 256, 0, 0);
            __builtin_prefetch(wptr + k + 256, 0, 0);
        }
        v2f a = *(const v2f*)(aptr + k);
        v2f b = *(const v2f*)(wptr + k);
        acc = __builtin_amdgcn_wmma_f32_16x16x4_f32(
                  /*neg_a=*/false, a, /*neg_b=*/false, b,
                  /*c_mod=*/(short)0, acc, /*reuse_a=*/false, /*reuse_b=*/false);
    }

#pragma unroll
    for (int j = 0; j < 8; ++j)
        C[(size_t)(m0 + j + 8 * lh) * NN + (n0 + lm)] = acc[j];
}

// ---------------------------------------------------------------------------
// Strict sequential scan:  out[s] = tanh(pre[s] + Whh @ out[s-1]),  out[-1]=0.
//
// Persistent, grid-synchronized: 128 WGs x 8 rows. Whh tile lives in LDS for
// the whole layer (loaded once). Each step: stage h (4 KB) to LDS, one wave
// per output row (conflict-free LDS), shfl_xor reduce, tanh, store, then a
// device-scope atomic arrive + spin barrier (monotonic counter, memset to 0
// by the host-side launcher before every launch -> graph-replay safe).
// ---------------------------------------------------------------------------
#define RROWS 8
__global__ __launch_bounds__(256) void rnn_recurrence(
        const float* __restrict__ pre,   // [S][N]
        const float* __restrict__ Whh,   // [N][N]
        float* __restrict__ out,         // [S][N] (written in place, seq buffer)
        unsigned* __restrict__ syncctr,
        int nwg)
{
    __shared__ float whh_s[RROWS][NN];   // 32 KB
    __shared__ float h_s[NN];            //  4 KB   (LDS budget: 36/320 KB)

    int tid  = threadIdx.x;
    int row  = tid >> 5;                 // wave id == local row (wave32)
    int lane = tid & 31;
    int row0 = blockIdx.x * RROWS;

    for (int e = tid; e < RROWS * NN; e += 256)
        whh_s[e >> 10][e & (NN - 1)] = Whh[(size_t)(row0 + (e >> 10)) * NN + (e & (NN - 1))];
    __syncthreads();

    for (int s = 0; s < SEQ; ++s) {
        float sum = 0.0f;
        if (s > 0) {                                     // h_{-1} == 0
            const float* hrow = out + (size_t)(s - 1) * NN;
            for (int j = tid; j < NN; j += 256) h_s[j] = hrow[j];
            __syncthreads();
            const float* wr = whh_s[row];
#pragma unroll 8
            for (int i = lane; i < NN; i += 32)          // lanes read consecutive
                sum += wr[i] * h_s[i];                   // LDS words: conflict-free
        }
#pragma unroll
        for (int off = 16; off > 0; off >>= 1)
            sum += __shfl_xor(sum, off, 32);

        if (lane == 0) {
            int r = row0 + row;
            out[(size_t)s * NN + r] = tanhf(pre[(size_t)s * NN + r] + sum);
        }

        // ---- device-wide step barrier ----
        __threadfence();
        __syncthreads();
        if (tid == 0) {
            __hip_atomic_fetch_add(syncctr, 1u, __ATOMIC_ACQ_REL, __HIP_MEMORY_SCOPE_AGENT);
            unsigned target = (unsigned)nwg * (unsigned)(s + 1);
            while (__hip_atomic_load(syncctr, __ATOMIC_ACQUIRE, __HIP_MEMORY_SCOPE_AGENT) < target)
                __builtin_amdgcn_s_sleep(1);
        }
        __syncthreads();
    }
}

// ---------------------------------------------------------------------------
// Orchestration (all stream-ordered, graph-capture safe).
// d_ws layout: seqA(4MB) | seqB(4MB) | pre(4MB) | sync counter
// ---------------------------------------------------------------------------
extern "C" void kernel_launch(void* const* d_in, const int* in_sizes, int n_in,
                              void* d_out, int out_size, void* d_ws, size_t ws_size,
                              hipStream_t stream) {
    const float* x    = (const float*)d_in[0];
    const float* Wih  = (const float*)d_in[1];
    const float* Whh  = (const float*)d_in[2];
    const float* bih  = (const float*)d_in[3];
    const float* bhh  = (const float*)d_in[4];
    const float* linW = (const float*)d_in[5];
    const float* linb = (const float*)d_in[6];
    float* out = (float*)d_out;

    const size_t matBytes = (size_t)SEQ * NN * sizeof(float);
    float*    seqA = (float*)d_ws;
    float*    seqB = (float*)((char*)d_ws + matBytes);
    float*    pre  = (float*)((char*)d_ws + 2 * matBytes);
    unsigned* sctr = (unsigned*)((char*)d_ws + 3 * matBytes);

    const dim3 gemmGrid(8, 64);                 // (N/128, M/16); 8 waves/WG on N
    const int  flatBlocks = (SEQ * NN) / 256;   // 4096

    rnn_extract_x<<<flatBlocks, 256, 0, stream>>>(x, seqA);

    float* cur = seqA;
    float* nxt = seqB;
    for (int t = 0; t < TT; ++t) {
        for (int l = 0; l < LL; ++l) {
            const size_t wOff = (size_t)(t * LL + l) * NN * NN;
            const size_t bOff = (size_t)(t * LL + l) * NN;
            // pre = cur @ Wih^T + bih + bhh   (WMMA fp32 GEMM)
            rnn_gemm_wmma_bias<<<gemmGrid, 256, 0, stream>>>(
                cur, Wih + wOff, bih + bOff, bhh + bOff, pre);
            // reset barrier counter (capture-legal), then sequential scan
            hipMemsetAsync(sctr, 0, sizeof(unsigned), stream);
            rnn_recurrence<<<128, 256, 0, stream>>>(pre, Whh + wOff, cur, sctr, 128);
        }
        // nxt = cur @ lin_W[t]^T + lin_b[t]
        rnn_gemm_wmma_bias<<<gemmGrid, 256, 0, stream>>>(
            cur, linW + (size_t)t * NN * NN, linb + (size_t)t * NN, nullptr, nxt);
        rnn_store_out<<<flatBlocks, 256, 0, stream>>>(nxt, out, t);
        float* tmp = cur; cur = nxt; nxt = tmp;
    }
}